// SCG_22368189678012
// MI455X (gfx1250) — compile-verified
//
#include <hip/hip_runtime.h>
#include <hip/hip_bf16.h>

// ---------------- problem constants ----------------
#define B_    4
#define H_    4096
#define W_    4096
#define NH_   2048
#define NW_   2048
#define NODES_ (NH_ * NW_)           // 4194304
#define PIX_   (B_ * NH_ * NW_)      // 16777216 (gx / mu / lv / z elements)
#define ASZ_   (B_ * NH_ * NH_)      // 16777216 (A / LA elements)

// WMMA / TDM vector types
typedef __attribute__((ext_vector_type(2))) float        v2f;
typedef __attribute__((ext_vector_type(8))) float        v8f;
typedef __attribute__((ext_vector_type(4))) unsigned int u32x4;
typedef __attribute__((ext_vector_type(4))) int          i32x4;
typedef __attribute__((ext_vector_type(8))) int          i32x8;

// ---------------- 1) 2x2 average pool ----------------
__launch_bounds__(256)
__global__ void pool_kernel(const float* __restrict__ x, float* __restrict__ gx) {
    int gid = blockIdx.x * 256 + threadIdx.x;            // 0 .. PIX_-1
    int j = gid & (NW_ - 1);
    int i = (gid >> 11) & (NH_ - 1);
    int b = gid >> 22;
    size_t base = (size_t)b * H_ * W_ + (size_t)(2 * i) * W_ + (size_t)(2 * j);
    float2 t = *(const float2*)(x + base);
    float2 u = *(const float2*)(x + base + W_);
    gx[gid] = 0.25f * (t.x + t.y + u.x + u.y);
}

// ---------------- 2) conv3x3 -> mu, affine -> lv, z = mu+exp(lv), KL partials ----------------
__launch_bounds__(256)
__global__ void conv_z_kernel(const float* __restrict__ gx,
                              const float* __restrict__ w_mu, const float* __restrict__ b_mu,
                              const float* __restrict__ w_lv, const float* __restrict__ b_lv,
                              float* __restrict__ zbuf, float* __restrict__ klpart) {
    int gid = blockIdx.x * 256 + threadIdx.x;
    int j = gid & (NW_ - 1);
    int i = (gid >> 11) & (NH_ - 1);
    int b = gid >> 22;
    const float* g = gx + (size_t)b * NH_ * NW_;

    float mu = b_mu[0];
#pragma unroll
    for (int kh = 0; kh < 3; ++kh) {
        int ii = i + kh - 1;
        if (ii < 0 || ii >= NH_) continue;
#pragma unroll
        for (int kw = 0; kw < 3; ++kw) {
            int jj = j + kw - 1;
            if (jj < 0 || jj >= NW_) continue;
            mu += w_mu[kh * 3 + kw] * g[(size_t)ii * NW_ + jj];
        }
    }
    float gc  = g[(size_t)i * NW_ + j];
    float lv  = w_lv[0] * gc + b_lv[0];
    float elv = expf(lv);
    zbuf[gid] = mu + elv;

    // KL term: 1 + 2*lv - mu^2 - exp(lv)^2
    float t = 1.0f + 2.0f * lv - mu * mu - elv * elv;
    __shared__ float red[256];
    int tid = threadIdx.x;
    red[tid] = t;
    __syncthreads();
    for (int off = 128; off > 0; off >>= 1) {
        if (tid < off) red[tid] += red[tid + off];
        __syncthreads();
    }
    if (tid == 0) klpart[blockIdx.x] = red[0];
}

// ---------------- 3) Gram: A = relu(z z^T), TDM-staged, double-buffered WMMA ----------------
#define KC_    32
#define LDP_   36                         // padded LDS row stride (floats), produced by TDM pad
#define TILEF_ (128 * LDP_)               // floats per staged tile
#define TILEB_ (TILEF_ * 4)               // bytes per staged tile

// Issue one TDM 2D tile load: 128 rows x 32 floats from a 2048x2048 f32 tensor,
// with 4-DWORD padding after every 32 DWORDs (row stride 36 floats in LDS).
__device__ __forceinline__ void tdm_load_tile(unsigned lds_off, unsigned long long gaddr) {
    u32x4 g0;
    g0.x = 1u;                                          // count=1 (valid user descriptor)
    g0.y = lds_off;                                     // lds_addr (bytes)
    g0.z = (unsigned)(gaddr & 0xFFFFFFFFull);           // global_addr[31:0]
    g0.w = (unsigned)(gaddr >> 32) | (2u << 30);        // global_addr[56:32] | type=2

    union { i32x8 v; unsigned d[8]; } g1;
    // dw0: wg_mask=0 | data_size=2(4B)<<16 | pad_en<<20 | pad_interval=4(32DW)<<22 | pad_amount=3(4DW)<<25
    g1.d[0] = (2u << 16) | (1u << 20) | (4u << 22) | (3u << 25);
    // dw1: atomic_barrier_addr=0 | tensor_dim0.lo16 << 16       (tensor_dim0 = 2048)
    g1.d[1] = (2048u & 0xFFFFu) << 16;
    // dw2: tensor_dim0.hi16 | tensor_dim1.lo16 << 16            (tensor_dim1 = 2048)
    g1.d[2] = (2048u >> 16) | ((2048u & 0xFFFFu) << 16);
    // dw3: tensor_dim1.hi16 | tile_dim0 << 16                   (tile_dim0 = 32)
    g1.d[3] = (2048u >> 16) | (32u << 16);
    // dw4: tile_dim1 | tile_dim2 << 16                          (tile_dim1 = 128)
    g1.d[4] = 128u;
    // dw5: tensor_dim0_stride[31:0]                             (stride = 2048 elements)
    g1.d[5] = 2048u;
    // dw6: tensor_dim0_stride[47:32] | tensor_dim1_stride.lo16 << 16
    g1.d[6] = 0u;
    // dw7: tensor_dim1_stride[47:16]
    g1.d[7] = 0u;

    i32x4 zz = {0, 0, 0, 0};
#if __clang_major__ >= 23
    i32x8 z8 = {0, 0, 0, 0, 0, 0, 0, 0};
    __builtin_amdgcn_tensor_load_to_lds(g0, g1.v, zz, zz, z8, 0);
#else
    __builtin_amdgcn_tensor_load_to_lds(g0, g1.v, zz, zz, 0);
#endif
}

__launch_bounds__(128)
__global__ void gram_wmma_kernel(const float* __restrict__ zbuf, float* __restrict__ Aout) {
    __shared__ float lds[2 * 2 * TILEF_];   // [buf][A/B][128*36] = 73,728 B

    const int b      = blockIdx.z;
    const int rowBlk = blockIdx.y * 128;
    const int colBlk = blockIdx.x * 128;
    const float* z   = zbuf + (size_t)b * NH_ * NW_;
    const float* zA  = z + (size_t)rowBlk * NW_;
    const float* zB  = z + (size_t)colBlk * NW_;

    const int tid   = threadIdx.x;
    const int lane  = tid & 31;
    const int wave  = tid >> 5;
    const int wr    = (wave >> 1) * 64;     // wave's row offset in 128x128 tile
    const int wc    = (wave & 1) * 64;      // wave's col offset
    const int lrow  = lane & 15;            // M / N index within 16
    const int khalf = (lane >> 4) * 2;      // K sub-offset 0 or 2

    v8f acc[4][4];
#pragma unroll
    for (int i = 0; i < 4; ++i)
#pragma unroll
        for (int j = 0; j < 4; ++j) acc[i][j] = v8f{};

    const unsigned ldsBase = (unsigned)(uintptr_t)lds;

    // prologue: stage chunk 0 into buffer 0
    if (wave == 0) {
        tdm_load_tile(ldsBase,          (unsigned long long)(uintptr_t)zA);
        tdm_load_tile(ldsBase + TILEB_, (unsigned long long)(uintptr_t)zB);
    }

    const int NCH = NW_ / KC_;   // 64 chunks
    for (int c = 0; c < NCH; ++c) {
        if (wave == 0) __builtin_amdgcn_s_wait_tensorcnt(0);
        __syncthreads();   // chunk c resident; all waves done reading buf[(c+1)&1] (from chunk c-1)

        if (wave == 0 && c + 1 < NCH) {
            unsigned nb = (unsigned)((c + 1) & 1) * 2u * (unsigned)TILEB_;
            tdm_load_tile(ldsBase + nb,          (unsigned long long)(uintptr_t)(zA + (c + 1) * KC_));
            tdm_load_tile(ldsBase + nb + TILEB_, (unsigned long long)(uintptr_t)(zB + (c + 1) * KC_));
        }

        const float* tA = lds + (c & 1) * 2 * TILEF_;
        const float* tB = tA + TILEF_;

#pragma unroll
        for (int ks = 0; ks < KC_; ks += 4) {
            v2f af[4], bf[4];
#pragma unroll
            for (int t = 0; t < 4; ++t) {
                af[t] = *(const v2f*)(tA + (wr + t * 16 + lrow) * LDP_ + ks + khalf);
                bf[t] = *(const v2f*)(tB + (wc + t * 16 + lrow) * LDP_ + ks + khalf);
            }
#pragma unroll
            for (int i = 0; i < 4; ++i)
#pragma unroll
                for (int j = 0; j < 4; ++j)
                    acc[i][j] = __builtin_amdgcn_wmma_f32_16x16x4_f32(
                        false, af[i], false, bf[j], (short)0, acc[i][j], false, false);
        }
    }

    // write out with ReLU.  C/D layout: VGPR v, lanes 0-15 -> M=v, lanes 16-31 -> M=v+8
    float* Ab = Aout + (size_t)b * NH_ * NH_;
#pragma unroll
    for (int i = 0; i < 4; ++i)
#pragma unroll
        for (int j = 0; j < 4; ++j) {
            int row0 = rowBlk + wr + i * 16 + (lane >> 4) * 8;
            int col  = colBlk + wc + j * 16 + lrow;
            float* outp = Ab + (size_t)row0 * NH_ + col;
#pragma unroll
            for (int v = 0; v < 8; ++v) {
                float val = acc[i][j][v];
                outp[(size_t)v * NH_] = val > 0.0f ? val : 0.0f;
            }
        }
}

// ---------------- 4) per-batch diagonal stats: gama[b], dl partial ----------------
__launch_bounds__(256)
__global__ void diag_stats_kernel(const float* __restrict__ A,
                                  float* __restrict__ gama, float* __restrict__ dls) {
    int b = blockIdx.x, tid = threadIdx.x;
    const float* Ab = A + (size_t)b * NH_ * NH_;
    float s = 0.0f, d = 0.0f;
    for (int n = tid; n < NH_; n += 256) {
        float a = Ab[(size_t)n * NH_ + n];
        s += a;
        if (a < 1.0f) d += logf(a + 1e-7f);
    }
    __shared__ float sa[256], sb[256];
    sa[tid] = s; sb[tid] = d;
    __syncthreads();
    for (int off = 128; off > 0; off >>= 1) {
        if (tid < off) { sa[tid] += sa[tid + off]; sb[tid] += sb[tid + off]; }
        __syncthreads();
    }
    if (tid == 0) {
        float mean = sa[0] / (float)NH_;
        gama[b] = sqrtf(1.0f + 1.0f / mean);
        dls[b]  = sb[0];
    }
}

// ---------------- 5) finalize scalars: gmean, loss ----------------
__launch_bounds__(256)
__global__ void finalize_kernel(const float* __restrict__ klpart,
                                const float* __restrict__ gama, const float* __restrict__ dls,
                                float* __restrict__ loss_out, float* __restrict__ gmean_out) {
    int tid = threadIdx.x;
    float s = 0.0f;
    for (int i = tid; i < 65536; i += 256) s += klpart[i];
    __shared__ float sm[256];
    sm[tid] = s;
    __syncthreads();
    for (int off = 128; off > 0; off >>= 1) {
        if (tid < off) sm[tid] += sm[tid + off];
        __syncthreads();
    }
    if (tid == 0) {
        float klsum = sm[0];
        float gmean = 0.25f * (gama[0] + gama[1] + gama[2] + gama[3]);
        float dlsum = dls[0] + dls[1] + dls[2] + dls[3];
        float kl = (-0.5f / (float)NODES_) * (klsum / (float)PIX_);
        float dl = gmean * dlsum / (float)ASZ_;
        loss_out[0]  = kl - dl;
        gmean_out[0] = gmean;
    }
}

// ---------------- 6) diag update: A[ii] += gama[b]*Ad + 1 ----------------
__launch_bounds__(256)
__global__ void diag_update_kernel(float* __restrict__ A, const float* __restrict__ gama) {
    int idx = blockIdx.x * 256 + threadIdx.x;   // 0 .. B*NH-1
    int b = idx >> 11, n = idx & (NH_ - 1);
    float* p = A + (size_t)b * NH_ * NH_ + (size_t)n * NH_ + n;
    float a = *p;
    *p = a + gama[b] * a + 1.0f;
}

// ---------------- 7) degree: dinv[b][m] = (sum_n A[b][n][m] + eps)^-0.5 ----------------
__launch_bounds__(256)
__global__ void row_sums_kernel(const float* __restrict__ A, float* __restrict__ dinv) {
    int idx = blockIdx.x * 256 + threadIdx.x;   // 0 .. B*NH-1
    int b = idx >> 11, m = idx & (NH_ - 1);
    const float* Ab = A + (size_t)b * NH_ * NH_;
    float s = 0.0f;
    for (int n = 0; n < NH_; ++n) s += Ab[(size_t)n * NH_ + m];
    dinv[idx] = 1.0f / sqrtf(s + 1e-5f);
}

// ---------------- 8) LA = D^-1/2 A D^-1/2 (in place, float4) ----------------
__launch_bounds__(256)
__global__ void normalize_kernel(float* __restrict__ A, const float* __restrict__ dinv) {
    int idx4 = blockIdx.x * 256 + threadIdx.x;  // 0 .. ASZ_/4-1
    size_t e = (size_t)idx4 * 4;
    int b = (int)(e >> 22);
    int r = (int)((e >> 11) & (NH_ - 1));
    int c = (int)(e & (NH_ - 1));
    float dn = dinv[b * NH_ + r];
    float4 dm = *(const float4*)(dinv + b * NH_ + c);
    float4 v  = *(float4*)(A + e);
    v.x *= dn * dm.x; v.y *= dn * dm.y; v.z *= dn * dm.z; v.w *= dn * dm.w;
    *(float4*)(A + e) = v;
}

// ---------------- 9) z_hat = gmean * mu * (1 - lv)  (recompute mu, lv from gx) ----------------
__launch_bounds__(256)
__global__ void zhat_kernel(const float* __restrict__ gx,
                            const float* __restrict__ w_mu, const float* __restrict__ b_mu,
                            const float* __restrict__ w_lv, const float* __restrict__ b_lv,
                            const float* __restrict__ gmean_p, float* __restrict__ zhat) {
    int gid = blockIdx.x * 256 + threadIdx.x;
    int j = gid & (NW_ - 1);
    int i = (gid >> 11) & (NH_ - 1);
    int b = gid >> 22;
    const float* g = gx + (size_t)b * NH_ * NW_;

    float mu = b_mu[0];
#pragma unroll
    for (int kh = 0; kh < 3; ++kh) {
        int ii = i + kh - 1;
        if (ii < 0 || ii >= NH_) continue;
#pragma unroll
        for (int kw = 0; kw < 3; ++kw) {
            int jj = j + kw - 1;
            if (jj < 0 || jj >= NW_) continue;
            mu += w_mu[kh * 3 + kw] * g[(size_t)ii * NW_ + jj];
        }
    }
    float lv = w_lv[0] * g[(size_t)i * NW_ + j] + b_lv[0];
    zhat[gid] = gmean_p[0] * mu * (1.0f - lv);
}

// ---------------- launch ----------------
extern "C" void kernel_launch(void* const* d_in, const int* in_sizes, int n_in,
                              void* d_out, int out_size, void* d_ws, size_t ws_size,
                              hipStream_t stream) {
    (void)in_sizes; (void)n_in; (void)out_size; (void)ws_size;
    const float* x    = (const float*)d_in[0];
    const float* w_mu = (const float*)d_in[1];
    const float* b_mu = (const float*)d_in[2];
    const float* w_lv = (const float*)d_in[3];
    const float* b_lv = (const float*)d_in[4];

    float* out   = (float*)d_out;
    float* LA    = out;                                   // ASZ_ floats
    float* gx    = out + (size_t)ASZ_;                    // PIX_ floats
    float* loss  = gx + (size_t)PIX_;                     // 1 float
    float* zhat  = loss + 1;                              // PIX_ floats

    float* ws     = (float*)d_ws;
    float* gmeanp = ws + 1;
    float* gama   = ws + 2;        // 4
    float* dls    = ws + 8;        // 4
    float* dinv   = ws + 16;       // B*NH = 8192
    float* klpart = ws + 8320;     // 65536
    float* zbuf   = ws + 81920;    // PIX_ floats

    const int NB = PIX_ / 256;     // 65536

    pool_kernel  <<<NB, 256, 0, stream>>>(x, gx);
    conv_z_kernel<<<NB, 256, 0, stream>>>(gx, w_mu, b_mu, w_lv, b_lv, zbuf, klpart);

    dim3 ggrid(NH_ / 128, NH_ / 128, B_);                 // 16 x 16 x 4
    gram_wmma_kernel<<<ggrid, 128, 0, stream>>>(zbuf, LA);

    diag_stats_kernel <<<B_, 256, 0, stream>>>(LA, gama, dls);
    finalize_kernel   <<<1, 256, 0, stream>>>(klpart, gama, dls, loss, gmeanp);
    diag_update_kernel<<<(B_ * NH_) / 256, 256, 0, stream>>>(LA, gama);
    row_sums_kernel   <<<(B_ * NH_) / 256, 256, 0, stream>>>(LA, dinv);
    normalize_kernel  <<<ASZ_ / 4 / 256, 256, 0, stream>>>(LA, dinv);
    zhat_kernel       <<<NB, 256, 0, stream>>>(gx, w_mu, b_mu, w_lv, b_lv, gmeanp, zhat);
}